// ChainGrowSolver_34754875359397
// MI455X (gfx1250) — compile-verified
//
#include <hip/hip_runtime.h>
#include <stdint.h>

// pts_world[i] = R[link[i]] * pts[i] + t[link[i]];  out = (pts_world f32[3N], link_ids i32[N])
//
// Roofline: 32 B/point moved, 18 FLOP/point -> 0.56 FLOP/byte; 268 MB @ 23.3 TB/s
// ~= 11.5 us floor, ~10x memory-bound. Optimized path on MI455X:
//  - 64 transforms (4 KB) staged to LDS per block via gfx1250 async copy
//    (global_load_async_to_lds_b128, ASYNCcnt, s_wait_asynccnt), gathered with
//    ds_load_b128 per point (only reused data lives in LDS).
//  - 4 points / thread / iter: xyz stream moves as 3x b128 loads + 3x b128
//    stores (16B-aligned), ids as one b128 each way -> minimal VMEM issue/byte.
//  - Non-temporal hints on all streaming traffic (no reuse, > L2 capacity),
//    global_prefetch_b8 for the next grid-stride tile.

typedef float f32x4 __attribute__((ext_vector_type(4)));
typedef int   i32x4 __attribute__((ext_vector_type(4)));

#define NLINKS 64

__global__ __launch_bounds__(256) void ChainGrowSolver_xform_kernel(
    const float* __restrict__ pts,         // (N,3) f32
    const float* __restrict__ transforms,  // (64,4,4) f32 row-major
    const int*   __restrict__ link_ids,    // (N,) i32
    float*       __restrict__ out_pts,     // (N,3) f32
    int*         __restrict__ out_ids,     // (N,) i32
    int n)
{
  // 64 transforms * 16 floats = 4 KB staged in LDS (row-major 4x4 per link).
  __shared__ __attribute__((aligned(16))) float smem[NLINKS * 16];

  // Cooperative async stage: 4096 B as 256 x 16 B, one chunk per thread.
  {
    unsigned c = threadIdx.x;  // 0..255
    uint64_t gaddr   = (uint64_t)(uintptr_t)transforms + (uint64_t)c * 16u;
    unsigned lds_off = (unsigned)(uintptr_t)(&smem[0]) + c * 16u;  // low 32b of flat = LDS offset
    asm volatile("global_load_async_to_lds_b128 %0, %1, off"
                 :: "v"(lds_off), "v"(gaddr)
                 : "memory");
    asm volatile("s_wait_asynccnt 0x0" ::: "memory");
  }
  __syncthreads();

  const int gtid    = (int)(blockIdx.x * blockDim.x + threadIdx.x);
  const int stride  = (int)(blockDim.x * gridDim.x);
  const int ngroups = n >> 2;                 // groups of 4 points

  for (int g = gtid; g < ngroups; g += stride) {
    const size_t base = (size_t)g * 4u;       // first point in group

    // Prefetch the next tile of the point stream (global_prefetch_b8).
    if (g + stride < ngroups) {
      __builtin_prefetch(pts + ((size_t)(g + stride) * 4u) * 3u, 0, 1);
    }

    // 4 points = 12 floats = 3 x b128 (48 B, 16-byte aligned).
    const f32x4* pv = (const f32x4*)(pts + base * 3u);
    const f32x4 a = __builtin_nontemporal_load(pv + 0);
    const f32x4 b = __builtin_nontemporal_load(pv + 1);
    const f32x4 c = __builtin_nontemporal_load(pv + 2);
    const i32x4 id4 = __builtin_nontemporal_load((const i32x4*)(link_ids + base));

    const float px[4] = {a.x, a.w, b.z, c.y};
    const float py[4] = {a.y, b.x, b.w, c.z};
    const float pz[4] = {a.z, b.y, c.x, c.w};
    const int   idv[4] = {id4.x, id4.y, id4.z, id4.w};

    float wx[4], wy[4], wz[4];
#pragma unroll
    for (int k = 0; k < 4; ++k) {
      // Gather this point's transform rows from LDS (3 x ds_load_b128).
      const f32x4* T = (const f32x4*)(&smem[idv[k] * 16]);
      const f32x4 r0 = T[0];   // R00 R01 R02 t0
      const f32x4 r1 = T[1];   // R10 R11 R12 t1
      const f32x4 r2 = T[2];   // R20 R21 R22 t2
      wx[k] = fmaf(r0.x, px[k], fmaf(r0.y, py[k], fmaf(r0.z, pz[k], r0.w)));
      wy[k] = fmaf(r1.x, px[k], fmaf(r1.y, py[k], fmaf(r1.z, pz[k], r1.w)));
      wz[k] = fmaf(r2.x, px[k], fmaf(r2.y, py[k], fmaf(r2.z, pz[k], r2.w)));
    }

    // Repack 4 transformed points into 3 x b128 stores.
    const f32x4 oa = {wx[0], wy[0], wz[0], wx[1]};
    const f32x4 ob = {wy[1], wz[1], wx[2], wy[2]};
    const f32x4 oc = {wz[2], wx[3], wy[3], wz[3]};
    f32x4* ov = (f32x4*)(out_pts + base * 3u);
    __builtin_nontemporal_store(oa, ov + 0);
    __builtin_nontemporal_store(ob, ov + 1);
    __builtin_nontemporal_store(oc, ov + 2);
    __builtin_nontemporal_store(id4, (i32x4*)(out_ids + base));
  }

  // Scalar tail for n % 4 (N = 8388608 is divisible; stay correct anyway).
  for (int i = (n & ~3) + gtid; i < n; i += stride) {
    const int id = link_ids[i];
    const float x = pts[(size_t)i * 3u + 0];
    const float y = pts[(size_t)i * 3u + 1];
    const float z = pts[(size_t)i * 3u + 2];
    const f32x4* T = (const f32x4*)(&smem[id * 16]);
    const f32x4 r0 = T[0], r1 = T[1], r2 = T[2];
    out_pts[(size_t)i * 3u + 0] = fmaf(r0.x, x, fmaf(r0.y, y, fmaf(r0.z, z, r0.w)));
    out_pts[(size_t)i * 3u + 1] = fmaf(r1.x, x, fmaf(r1.y, y, fmaf(r1.z, z, r1.w)));
    out_pts[(size_t)i * 3u + 2] = fmaf(r2.x, x, fmaf(r2.y, y, fmaf(r2.z, z, r2.w)));
    out_ids[i] = id;
  }
}

extern "C" void kernel_launch(void* const* d_in, const int* in_sizes, int n_in,
                              void* d_out, int out_size, void* d_ws, size_t ws_size,
                              hipStream_t stream) {
  (void)n_in; (void)d_ws; (void)ws_size; (void)out_size;

  const float* pts        = (const float*)d_in[0];  // 3N f32
  const float* transforms = (const float*)d_in[1];  // 64*16 f32
  const int*   link_ids   = (const int*)d_in[2];    // N i32

  const int n = in_sizes[0] / 3;

  float* out_pts = (float*)d_out;                          // first 3N floats
  int*   out_ids = (int*)((float*)d_out + (size_t)n * 3);  // then N int32 ids

  const int threads = 256;
  const int ngroups = n / 4;
  // ~4 groups (16 points) per thread; plenty of blocks to cover every WGP.
  int blocks = (ngroups + threads * 4 - 1) / (threads * 4);
  if (blocks < 1) blocks = 1;

  ChainGrowSolver_xform_kernel<<<blocks, threads, 0, stream>>>(
      pts, transforms, link_ids, out_pts, out_ids, n);
}